// ContinualSVGP_9019431321735
// MI455X (gfx1250) — compile-verified
//
#include <hip/hip_runtime.h>

// Problem constants (from reference): N=32768, M=64, D=8, O=4, H=4
#define HDIM 4
#define ODIM 4
#define JITTER 1e-4f

// workspace layout per (h,o) pair, in floats
#define WS_LINV 0        // 64*64
#define WS_B    4096     // 64*64
#define WS_ZS   8192     // 64*8
#define WS_Z2   8704     // 64
#define WS_Q    8768     // 64
#define WS_ILS  8832     // 8
#define WS_VAR  8840     // 1
#define WS_STRIDE 8960

typedef __attribute__((ext_vector_type(2))) float v2f;
typedef __attribute__((ext_vector_type(8))) float v8f;

// ---------------------------------------------------------------------------
// Phase 1: per (h,o) pair, build Linv = chol(kuu)^-1, B = A^T Linv, q, etc.
// One 64-thread block per pair (16 blocks). All tiny f32 VALU work.
// ---------------------------------------------------------------------------
__global__ __launch_bounds__(64) void svgp_prep(
    const float* __restrict__ z, const float* __restrict__ u_mean,
    const float* __restrict__ u_tril_vec, const float* __restrict__ log_ls,
    const float* __restrict__ log_var, float* __restrict__ ws)
{
    __shared__ float sL[64 * 64];
    __shared__ float sLinv[64 * 64];
    __shared__ float sA[64 * 64];
    __shared__ float sZs[64 * 8];
    __shared__ float sZ2[64];
    __shared__ float sT[64];

    const int p   = blockIdx.x;        // pair = h*ODIM + o
    const int o   = p % ODIM;
    const int tid = threadIdx.x;       // 0..63
    float* wp = ws + (size_t)p * WS_STRIDE;

    float ils[8];
#pragma unroll
    for (int d = 0; d < 8; ++d) ils[d] = __expf(-log_ls[p * 8 + d]);
    const float var = __expf(log_var[p]);

    // zs = z / ls, z2 = ||zs||^2  (thread tid == inducing index m)
    {
        float s = 0.f;
#pragma unroll
        for (int d = 0; d < 8; ++d) {
            float v = z[(o * 64 + tid) * 8 + d] * ils[d];
            sZs[tid * 8 + d] = v;
            s += v * v;
        }
        sZ2[tid] = s;
    }
    __syncthreads();

    // kuu row i = tid  (var * exp(-0.5 * sqdist) + jitter on diag)
    {
        const int i = tid;
        for (int j = 0; j < 64; ++j) {
            float dot = 0.f;
#pragma unroll
            for (int d = 0; d < 8; ++d) dot += sZs[i * 8 + d] * sZs[j * 8 + d];
            float v = var * __expf(-0.5f * (sZ2[i] + sZ2[j] - 2.f * dot));
            if (i == j) v += JITTER;
            sL[i * 64 + j] = v;
        }
    }
    __syncthreads();

    // Cholesky (in place in sL; lower triangle valid afterwards)
    for (int j = 0; j < 64; ++j) {
        if (tid == j) {
            float s = sL[j * 64 + j];
            for (int k = 0; k < j; ++k) s -= sL[j * 64 + k] * sL[j * 64 + k];
            sL[j * 64 + j] = sqrtf(s);
        }
        __syncthreads();
        if (tid > j) {
            float s = sL[tid * 64 + j];
            for (int k = 0; k < j; ++k) s -= sL[tid * 64 + k] * sL[j * 64 + k];
            sL[tid * 64 + j] = s / sL[j * 64 + j];
        }
        __syncthreads();
    }

    // Linv: forward-substitute column c = tid (columns are independent)
    {
        const int c = tid;
        for (int i = 0; i < c; ++i) sLinv[i * 64 + c] = 0.f;
        sLinv[c * 64 + c] = 1.f / sL[c * 64 + c];
        for (int i = c + 1; i < 64; ++i) {
            float s = 0.f;
            for (int k = c; k < i; ++k) s += sL[i * 64 + k] * sLinv[k * 64 + c];
            sLinv[i * 64 + c] = -s / sL[i * 64 + i];
        }
    }

    // A[i][j] = u_tril[i][j] / L[i][i]  (solve vs triu(L) == diag(L))
    {
        const int i    = tid;
        const float di = 1.f / sL[i * 64 + i];
        const int base = o * 2080 + i * (i + 1) / 2;
        for (int j = 0; j < 64; ++j)
            sA[i * 64 + j] = (j <= i) ? u_tril_vec[base + j] * di : 0.f;
    }
    __syncthreads();

    // t = Linv @ u_mean (row i = tid)
    {
        const int i = tid;
        float s = 0.f;
        for (int j = 0; j <= i; ++j) s += sLinv[i * 64 + j] * u_mean[o * 64 + j];
        sT[i] = s;
    }
    __syncthreads();

    // q[j] = sum_i Linv[i][j] * t[i]   ->  pred_mu = q . kfu_col
    {
        const int j = tid;
        float s = 0.f;
        for (int i = j; i < 64; ++i) s += sLinv[i * 64 + j] * sT[i];
        wp[WS_Q + j] = s;
    }

    // B[j][r] = sum_i A[i][j] * Linv[i][r]  (row j = tid), plus stash constants
    {
        const int j = tid;
        for (int r = 0; r < 64; ++r) {
            float s = 0.f;
            const int i0 = (j > r) ? j : r;   // A col j starts at i=j, Linv col r at i=r
            for (int i = i0; i < 64; ++i) s += sA[i * 64 + j] * sLinv[i * 64 + r];
            wp[WS_B + j * 64 + r] = s;
        }
        for (int r = 0; r < 64; ++r) wp[WS_LINV + j * 64 + r] = sLinv[j * 64 + r];
#pragma unroll
        for (int d = 0; d < 8; ++d) wp[WS_ZS + j * 8 + d] = sZs[j * 8 + d];
        wp[WS_Z2 + j] = sZ2[j];
        if (j < 8) wp[WS_ILS + j] = ils[j];
        if (j == 0) wp[WS_VAR] = var;
    }
}

// ---------------------------------------------------------------------------
// Phase 2: per pair, Y = Linv*K and W = B*K via V_WMMA_F32_16X16X4_F32,
// pred_mu = q.K, pred_var = var - ||Y||^2 + ||W||^2 per column.
// Block = 128 threads (4 waves); each wave owns a 64x16 kfu tile.
// ---------------------------------------------------------------------------
#define LSTR 65                 // padded LDS row stride (bank-conflict free)
#define COLS_PER_BLOCK 512

__global__ __launch_bounds__(128) void svgp_main(
    const float* __restrict__ x, const float* __restrict__ ws,
    float* __restrict__ out, int n, int blocksPerPair)
{
    __shared__ float sLinv[64 * LSTR];
    __shared__ float sB[64 * LSTR];
    __shared__ float sZs[64 * 8];
    __shared__ float sZ2[64];
    __shared__ float sQ[64];
    __shared__ float sK[4][64 * 16];
    __shared__ float sXs[4][16 * 8];
    __shared__ float sX2[4][16];

    const int pair  = blockIdx.x / blocksPerPair;
    const int chunk = blockIdx.x % blocksPerPair;
    const float* wp = ws + (size_t)pair * WS_STRIDE;
    const int tid  = threadIdx.x;
    const int wave = tid >> 5;
    const int lane = tid & 31;

    // stage per-pair matrices into LDS (padded rows)
    for (int i = tid; i < 4096; i += 128) {
        const int r = i >> 6, c = i & 63;
        sLinv[r * LSTR + c] = wp[WS_LINV + i];
        sB[r * LSTR + c]    = wp[WS_B + i];
    }
    for (int i = tid; i < 512; i += 128) sZs[i] = wp[WS_ZS + i];
    if (tid < 64) { sZ2[tid] = wp[WS_Z2 + tid]; sQ[tid] = wp[WS_Q + tid]; }
    float ils[8];
#pragma unroll
    for (int d = 0; d < 8; ++d) ils[d] = wp[WS_ILS + d];
    const float var = wp[WS_VAR];
    __syncthreads();

    const size_t varBase = (size_t)(HDIM * ODIM) * (size_t)n;  // pred_var offset
    const int nloc  = lane & 15;
    const int khalf = (lane >> 4) * 2;

    for (int it = 0; it < COLS_PER_BLOCK / 64; ++it) {
        const int col0 = chunk * COLS_PER_BLOCK + it * 64 + wave * 16;
        __syncthreads();

        // scaled query points for this wave's 16 columns
        if (lane < 16) {
            const int cidx = col0 + lane;
            float s = 0.f;
#pragma unroll
            for (int d = 0; d < 8; ++d) {
                float v = x[cidx * 8 + d] * ils[d];
                sXs[wave][lane * 8 + d] = v;
                s += v * v;
            }
            sX2[wave][lane] = s;
        }
        __syncthreads();

        // kfu tile: K[m][n] = var*exp(-0.5*(x2+z2-2*dot)), 32 entries per lane
        {
            const float xn2 = sX2[wave][nloc];
#pragma unroll 4
            for (int i = 0; i < 32; ++i) {
                const int m = 2 * i + (lane >> 4);
                float dot = 0.f;
#pragma unroll
                for (int d = 0; d < 8; ++d)
                    dot += sZs[m * 8 + d] * sXs[wave][nloc * 8 + d];
                sK[wave][m * 16 + nloc] =
                    var * __expf(-0.5f * (xn2 + sZ2[m] - 2.f * dot));
            }
        }
        __syncthreads();

        // dual GEMM: Y = Linv*K, W = B*K  (f32 WMMA, K-dim 64 = 16 chunks of 4)
        v8f accY[4] = {};
        v8f accW[4] = {};
#pragma unroll
        for (int kc = 0; kc < 16; ++kc) {
            const int k0 = kc * 4 + khalf;
            v2f bfrag;                        // B frag: lane n=L%16, k=k0+v
            bfrag.x = sK[wave][(k0 + 0) * 16 + nloc];
            bfrag.y = sK[wave][(k0 + 1) * 16 + nloc];
#pragma unroll
            for (int rt = 0; rt < 4; ++rt) {
                const int m = rt * 16 + nloc; // A frag: lane m=L%16, k=k0+v
                v2f aY; aY.x = sLinv[m * LSTR + k0]; aY.y = sLinv[m * LSTR + k0 + 1];
                accY[rt] = __builtin_amdgcn_wmma_f32_16x16x4_f32(
                    false, aY, false, bfrag, (short)0, accY[rt], false, false);
                v2f aW; aW.x = sB[m * LSTR + k0]; aW.y = sB[m * LSTR + k0 + 1];
                accW[rt] = __builtin_amdgcn_wmma_f32_16x16x4_f32(
                    false, aW, false, bfrag, (short)0, accW[rt], false, false);
            }
        }

        // per-column squared norms: lane holds col nloc, rows v + 8*(lane>>4)
        float d1 = 0.f, d2 = 0.f;
#pragma unroll
        for (int rt = 0; rt < 4; ++rt)
#pragma unroll
            for (int v = 0; v < 8; ++v) {
                d1 += accY[rt][v] * accY[rt][v];
                d2 += accW[rt][v] * accW[rt][v];
            }
        d1 += __shfl_xor(d1, 16, 32);   // combine row halves (lanes n, n+16)
        d2 += __shfl_xor(d2, 16, 32);

        if (lane < 16) {
            float mu = 0.f;
#pragma unroll 8
            for (int m = 0; m < 64; ++m) mu += sQ[m] * sK[wave][m * 16 + lane];
            const int col = col0 + lane;
            out[(size_t)pair * n + col]           = mu;
            out[varBase + (size_t)pair * n + col] = var - d1 + d2;
        }
    }
}

// ---------------------------------------------------------------------------
extern "C" void kernel_launch(void* const* d_in, const int* in_sizes, int n_in,
                              void* d_out, int out_size, void* d_ws, size_t ws_size,
                              hipStream_t stream)
{
    const float* x          = (const float*)d_in[0];
    const float* z          = (const float*)d_in[1];
    const float* u_mean     = (const float*)d_in[2];
    const float* u_tril_vec = (const float*)d_in[3];
    const float* log_ls     = (const float*)d_in[4];
    const float* log_var    = (const float*)d_in[5];
    float* out = (float*)d_out;
    float* ws  = (float*)d_ws;

    const int n = in_sizes[0] / 8;                 // N = 32768
    svgp_prep<<<HDIM * ODIM, 64, 0, stream>>>(z, u_mean, u_tril_vec,
                                              log_ls, log_var, ws);
    const int blocksPerPair = n / COLS_PER_BLOCK;  // 64
    svgp_main<<<HDIM * ODIM * blocksPerPair, 128, 0, stream>>>(
        x, ws, out, n, blocksPerPair);
}